// APIHyperInputLayer_79998060855349
// MI455X (gfx1250) — compile-verified
//
#include <hip/hip_runtime.h>

typedef float v2f __attribute__((ext_vector_type(2)));
typedef float v8f __attribute__((ext_vector_type(8)));

#define N_AGENTS  10
#define N_ENEMIES 11
#define ALLY_F    32
#define ENEMY_F   28
#define HYPER     64
#define OUT_F     256
#define BATCH     1024

#define KA   (HYPER * ALLY_F)              // 2048
#define KE   (HYPER * ENEMY_F)             // 1792
#define KTOT (KA + KE + ALLY_F + ENEMY_F)  // 3900
#define KPAD 3904                          // pad to multiple of 64

__device__ __forceinline__ v8f wmma_f32_16x16x4(v2f a, v2f b, v8f c) {
  // 8 args: (neg_a, A, neg_b, B, c_mod, C, reuse_a, reuse_b)
  return __builtin_amdgcn_wmma_f32_16x16x4_f32(false, a, false, b, (short)0, c,
                                               false, false);
}

// ---------------------------------------------------------------------------
// Kernel 1: h = relu(F @ W1 + b1) for both branches, WMMA f32 16x16x4.
// One wave per 16x16 output tile. Ally: 640x4 tiles, Enemy: 704x4 tiles.
// ---------------------------------------------------------------------------
__global__ void k_hidden(const float* __restrict__ fA, const float* __restrict__ fE,
                         const float* __restrict__ wa1, const float* __restrict__ ba1,
                         const float* __restrict__ we1, const float* __restrict__ be1,
                         float* __restrict__ hA, float* __restrict__ hE) {
  const int lane = threadIdx.x & 31;
  int w = blockIdx.x * (blockDim.x >> 5) + (threadIdx.x >> 5);
  const int allyTiles = (BATCH * N_AGENTS / 16) * (HYPER / 16);  // 2560

  const float *F, *W1, *B1;
  float* H;
  int K, ldF;
  if (w < allyTiles) {
    F = fA; W1 = wa1; B1 = ba1; H = hA; K = ALLY_F; ldF = ALLY_F;
  } else {
    w -= allyTiles;
    F = fE; W1 = we1; B1 = be1; H = hE; K = ENEMY_F; ldF = ENEMY_F;
  }
  const int row0 = (w >> 2) * 16;
  const int col0 = (w & 3) * 16;
  const int m  = lane & 15;          // M (for A) / N (for B, C)
  const int kb = (lane >> 4) * 2;    // K sub-offset per lane half

  v8f c = {};
  for (int k = 0; k < K; k += 4) {
    // (row*ldF + k + kb) is even for both ldF=32 and ldF=28 -> 8B aligned
    const v2f a = *(const v2f*)&F[(row0 + m) * ldF + k + kb];
    v2f b;
    b.x = W1[(k + kb)     * HYPER + col0 + m];
    b.y = W1[(k + kb + 1) * HYPER + col0 + m];
    c = wmma_f32_16x16x4(a, b, c);
  }
  const float bias = B1[col0 + m];
#pragma unroll
  for (int r = 0; r < 8; ++r) {
    const int rr = row0 + ((lane < 16) ? r : r + 8);
    H[rr * HYPER + col0 + m] = fmaxf(c[r] + bias, 0.0f);
  }
}

// ---------------------------------------------------------------------------
// Kernel 2: WcP = K-pair-interleaved W_cat. Logical W_cat[kk,o] rows:
//   [0,KA)        reshaped wa2   (kk = k*32 + f  -> wa2[k, f*256 + o])
//   [KA,KA+KE)    reshaped we2   (kk = k*28 + f  -> we2[k, f*256 + o])
//   then ba2 rows, be2 rows, zero pad to KPAD.
// Physical layout: WcP[p*512 + o*2 + j] = W_cat[2p + j, o]  (j in {0,1}),
// so a lane's {K,K+1} B-operand pair is one contiguous 8-byte load.
// ---------------------------------------------------------------------------
__device__ __forceinline__ float wcat_elem(int kk, int o,
                                           const float* __restrict__ wa2,
                                           const float* __restrict__ ba2,
                                           const float* __restrict__ we2,
                                           const float* __restrict__ be2) {
  if (kk < KA) {
    const int k = kk >> 5, f = kk & 31;
    return wa2[k * (ALLY_F * OUT_F) + f * OUT_F + o];
  } else if (kk < KA + KE) {
    const int t = kk - KA;
    const int k = t / ENEMY_F, f = t % ENEMY_F;
    return we2[k * (ENEMY_F * OUT_F) + f * OUT_F + o];
  } else if (kk < KA + KE + ALLY_F) {
    return ba2[(kk - (KA + KE)) * OUT_F + o];
  } else if (kk < KTOT) {
    return be2[(kk - (KA + KE + ALLY_F)) * OUT_F + o];
  }
  return 0.0f;
}

__global__ void k_repack(const float* __restrict__ wa2, const float* __restrict__ ba2,
                         const float* __restrict__ we2, const float* __restrict__ be2,
                         float* __restrict__ WcP) {
  const int p = blockIdx.x;    // K-pair row 0..KPAD/2-1
  const int o = threadIdx.x;   // 0..255
  v2f v;
  v.x = wcat_elem(2 * p,     o, wa2, ba2, we2, be2);
  v.y = wcat_elem(2 * p + 1, o, wa2, ba2, we2, be2);
  *(v2f*)&WcP[(size_t)p * (2 * OUT_F) + o * 2] = v;  // coalesced b64 store
}

// ---------------------------------------------------------------------------
// Kernel 3: Q[b, :] = entity-summed outer products (h ⊗ F), plus ΣF rows for
// the bias GEMM, plus zero pad. One block per batch element; entity data
// staged in LDS.
// ---------------------------------------------------------------------------
__global__ void k_buildq(const float* __restrict__ fA, const float* __restrict__ fE,
                         const float* __restrict__ hA, const float* __restrict__ hE,
                         float* __restrict__ Q) {
  __shared__ float s_hA[N_AGENTS * HYPER];      // 640
  __shared__ float s_fA[N_AGENTS * ALLY_F];     // 320
  __shared__ float s_hE[N_ENEMIES * HYPER];     // 704
  __shared__ float s_fE[N_ENEMIES * ENEMY_F];   // 308
  const int b = blockIdx.x;
  const int tid = threadIdx.x;
  for (int i = tid; i < N_AGENTS * HYPER; i += blockDim.x)
    s_hA[i] = hA[b * N_AGENTS * HYPER + i];
  for (int i = tid; i < N_AGENTS * ALLY_F; i += blockDim.x)
    s_fA[i] = fA[b * N_AGENTS * ALLY_F + i];
  for (int i = tid; i < N_ENEMIES * HYPER; i += blockDim.x)
    s_hE[i] = hE[b * N_ENEMIES * HYPER + i];
  for (int i = tid; i < N_ENEMIES * ENEMY_F; i += blockDim.x)
    s_fE[i] = fE[b * N_ENEMIES * ENEMY_F + i];
  __syncthreads();

  for (int kk = tid; kk < KPAD; kk += blockDim.x) {
    float s = 0.0f;
    if (kk < KA) {
      const int k = kk >> 5, f = kk & 31;
#pragma unroll
      for (int e = 0; e < N_AGENTS; ++e)
        s += s_hA[e * HYPER + k] * s_fA[e * ALLY_F + f];
    } else if (kk < KA + KE) {
      const int t = kk - KA;
      const int k = t / ENEMY_F, f = t % ENEMY_F;
#pragma unroll
      for (int e = 0; e < N_ENEMIES; ++e)
        s += s_hE[e * HYPER + k] * s_fE[e * ENEMY_F + f];
    } else if (kk < KA + KE + ALLY_F) {
      const int f = kk - (KA + KE);
#pragma unroll
      for (int e = 0; e < N_AGENTS; ++e) s += s_fA[e * ALLY_F + f];
    } else if (kk < KTOT) {
      const int f = kk - (KA + KE + ALLY_F);
#pragma unroll
      for (int e = 0; e < N_ENEMIES; ++e) s += s_fE[e * ENEMY_F + f];
    }
    Q[(size_t)b * KPAD + kk] = s;
  }
}

// ---------------------------------------------------------------------------
// Kernel 4: out [1024,256] = Q [1024,KPAD] @ W_cat, WMMA f32 16x16x4.
// Block tile 32x256 (8 waves; each wave a 32x32 tile -> 4 WMMAs per k-step).
// A tile staged in LDS via global_load_async_to_lds_b128 (ASYNCcnt path);
// stride 68 floats = 272 B keeps rows 16B-aligned and banks conflict-free
// (lane m -> bank 4m mod 64). B comes from the pair-interleaved WcP as b64.
// ---------------------------------------------------------------------------
#define GEMM_KC 64
#define SQ_LD   68
__global__ void k_gemm(const float* __restrict__ Q, const float* __restrict__ WcP,
                       float* __restrict__ Cout) {
  __shared__ float sQ[32 * SQ_LD];   // only LDS object -> offset 0
  const int row0 = blockIdx.x * 32;
  const int tid  = threadIdx.x;
  const int lane = tid & 31;
  const int n0   = (tid >> 5) * 32;  // wave -> 32-column slice
  const int m    = lane & 15;
  const int kb   = (lane >> 4) * 2;

  // Async-copy plan: 32 rows x 64 floats = 512 x b128 chunks, 2 per thread.
  const int ch0 = tid, ch1 = tid + 256;
  const int r0c = ch0 >> 4, q0c = (ch0 & 15) * 4;   // row, float-offset in row
  const int r1c = ch1 >> 4, q1c = (ch1 & 15) * 4;
  const unsigned lds0 = (unsigned)((r0c * SQ_LD + q0c) * 4);  // byte offsets
  const unsigned lds1 = (unsigned)((r1c * SQ_LD + q1c) * 4);
  const float* g0 = Q + (size_t)(row0 + r0c) * KPAD + q0c;
  const float* g1 = Q + (size_t)(row0 + r1c) * KPAD + q1c;

  v8f c00 = {}, c01 = {}, c10 = {}, c11 = {};
  for (int kc = 0; kc < KPAD; kc += GEMM_KC) {
    __syncthreads();  // previous chunk fully consumed before overwrite
    {
      const unsigned long long a0 = (unsigned long long)(g0 + kc);
      const unsigned long long a1 = (unsigned long long)(g1 + kc);
      asm volatile("global_load_async_to_lds_b128 %0, %1, off"
                   :: "v"(lds0), "v"(a0) : "memory");
      asm volatile("global_load_async_to_lds_b128 %0, %1, off"
                   :: "v"(lds1), "v"(a1) : "memory");
      asm volatile("s_wait_asynccnt 0x0" ::: "memory");
    }
    __syncthreads();  // all waves' async tiles landed in LDS
#pragma unroll 4
    for (int k = 0; k < GEMM_KC; k += 4) {
      const v2f a0 = *(const v2f*)&sQ[m * SQ_LD + k + kb];         // ds_load_b64
      const v2f a1 = *(const v2f*)&sQ[(16 + m) * SQ_LD + k + kb];
      const int kg = kc + k + kb;                                   // even
      const v2f b0 = *(const v2f*)&WcP[(size_t)(kg >> 1) * (2 * OUT_F) + (n0 + m) * 2];
      const v2f b1 = *(const v2f*)&WcP[(size_t)(kg >> 1) * (2 * OUT_F) + (n0 + 16 + m) * 2];
      c00 = wmma_f32_16x16x4(a0, b0, c00);
      c01 = wmma_f32_16x16x4(a0, b1, c01);
      c10 = wmma_f32_16x16x4(a1, b0, c10);
      c11 = wmma_f32_16x16x4(a1, b1, c11);
    }
  }
#pragma unroll
  for (int r = 0; r < 8; ++r) {
    const int rr = (lane < 16) ? r : r + 8;
    Cout[(row0 + rr) * OUT_F + n0 + m]           = c00[r];
    Cout[(row0 + rr) * OUT_F + n0 + 16 + m]      = c01[r];
    Cout[(row0 + 16 + rr) * OUT_F + n0 + m]      = c10[r];
    Cout[(row0 + 16 + rr) * OUT_F + n0 + 16 + m] = c11[r];
  }
}

// ---------------------------------------------------------------------------
extern "C" void kernel_launch(void* const* d_in, const int* in_sizes, int n_in,
                              void* d_out, int out_size, void* d_ws, size_t ws_size,
                              hipStream_t stream) {
  const float* fA  = (const float*)d_in[0];
  const float* fE  = (const float*)d_in[1];
  const float* wa1 = (const float*)d_in[2];
  const float* ba1 = (const float*)d_in[3];
  const float* wa2 = (const float*)d_in[4];
  const float* ba2 = (const float*)d_in[5];
  const float* we1 = (const float*)d_in[6];
  const float* be1 = (const float*)d_in[7];
  const float* we2 = (const float*)d_in[8];
  const float* be2 = (const float*)d_in[9];
  float* out = (float*)d_out;

  float* ws = (float*)d_ws;
  float* hA = ws;                               // 10240*64
  float* hE = hA + BATCH * N_AGENTS * HYPER;    // 11264*64
  float* Qm = hE + BATCH * N_ENEMIES * HYPER;   // 1024*KPAD (16B-aligned offset)
  float* Wc = Qm + (size_t)BATCH * KPAD;        // KPAD*256, pair-interleaved

  // waves: ally 2560 + enemy 2816 = 5376 tiles; 4 waves/block -> 1344 blocks
  k_hidden<<<1344, 128, 0, stream>>>(fA, fE, wa1, ba1, we1, be1, hA, hE);
  k_repack<<<KPAD / 2, OUT_F, 0, stream>>>(wa2, ba2, we2, be2, Wc);
  k_buildq<<<BATCH, 256, 0, stream>>>(fA, fE, hA, hE, Qm);
  k_gemm<<<BATCH / 32, 256, 0, stream>>>(Qm, Wc, out);
}